// DCG_88845693485562
// MI455X (gfx1250) — compile-verified
//
#include <hip/hip_runtime.h>
#include <cstdint>
#include <cstddef>

#define THREADS 256
constexpr int B_ = 512, N_ = 64, E_ = 512, A_ = 16, ITERS_ = 8;
constexpr int TE_ = 32;                  // edges per staged LDS tile (32 KB)
constexpr int NTILES_ = E_ / TE_;        // 16
constexpr float INV_N = 1.0f / (float)N_;
constexpr float INV_E = 1.0f / (float)E_;

typedef __attribute__((ext_vector_type(2))) float v2f;
typedef __attribute__((ext_vector_type(8))) float v8f;

struct __align__(16) Smem {
  float q [N_*A_];          // current beliefs
  float nv[N_*A_];          // node_vals copy
  float mf[E_*A_];          // forward messages
  float mb[E_*A_];          // backward messages
  float vf[TE_*A_];         // staged q[from]-mb for tile
  float vb[TE_*A_];         // staged q[to]-mf for tile
  float sf[TE_*A_];         // raw (uncentered) new mf tile
  float sb[TE_*A_];         // raw (uncentered) new mb tile
  float ebuf[2][TE_*A_*A_]; // double-buffered edge matrices (async dest)
  float red[THREADS];
  int   ef[E_];
  int   et[E_];
  int   a_cur[N_];
  int   a_best[N_];
  float qmax;
  int   upd;
};

// Issue one 32KB edge tile as async global->LDS B128 copies (8 instr/thread,
// lane-contiguous 16B chunks -> fully coalesced). Tracked by ASYNCcnt.
__device__ __forceinline__ void async_tile(const float* __restrict__ g,
                                           const float* l, int tid) {
  uint64_t ga = (uint64_t)(uintptr_t)g;
  uint32_t la = (uint32_t)(uintptr_t)l;   // low 32 bits of LDS aperture addr = LDS offset
#pragma unroll
  for (int c = 0; c < (TE_*A_*A_) / (THREADS*4); ++c) {  // 8 x b128 per thread
    uint32_t off = (uint32_t)(c*THREADS + tid) * 16u;
    asm volatile("global_load_async_to_lds_b128 %0, %1, off"
                 :: "v"(la + off), "v"(ga + (uint64_t)off) : "memory");
  }
}

// Mean-center a 16x16 message block: D = Src * (I - J/16), exact matmul,
// done as 4 chained V_WMMA_F32_16X16X4_F32 (one wave, EXEC all ones).
__device__ __forceinline__ void center16(const float* src, float* dst) {
  const int lane = threadIdx.x & 31;
  const int m  = lane & 15;     // row of A / col of B / col of D
  const int hi = lane >> 4;     // lane-half selects K pair
  v8f acc = {0.f,0.f,0.f,0.f,0.f,0.f,0.f,0.f};
#pragma unroll
  for (int j = 0; j < 4; ++j) {
    const int k0 = 4*j + 2*hi;
    v2f a; a.x = src[m*A_ + k0]; a.y = src[m*A_ + k0 + 1];     // A: 16x4 slice
    v2f bv;                                                     // B: rows k0,k0+1 of (I - J/16)
    bv.x = ((k0    ) == m ? 1.0f : 0.0f) - 0.0625f;
    bv.y = ((k0 + 1) == m ? 1.0f : 0.0f) - 0.0625f;
    acc = __builtin_amdgcn_wmma_f32_16x16x4_f32(false, a, false, bv,
                                                (short)0, acc, false, false);
  }
  const int mb0 = hi ? 8 : 0;   // D layout: VGPR r -> row r (+8 for upper lanes)
#pragma unroll
  for (int r = 0; r < 8; ++r) dst[(mb0 + r)*A_ + m] = acc[r];
}

__global__ void __launch_bounds__(THREADS)
dcg_maxsum_kernel(const float* __restrict__ node_vals,
                  const float* __restrict__ edge_vals,
                  const int* __restrict__ edges_from,
                  const int* __restrict__ edges_to,
                  float* __restrict__ out) {
  extern __shared__ char smem_raw[];
  Smem& s = *reinterpret_cast<Smem*>(smem_raw);
  const int tid = threadIdx.x;
  const int b   = blockIdx.x;
  const float* __restrict__ eg = edge_vals + (size_t)b * (size_t)(E_) * (A_*A_);

  // ---- init: beliefs, edge lists, zero messages ----
  for (int i = tid; i < N_*A_; i += THREADS) {
    float v = node_vals[(size_t)b*N_*A_ + i];
    s.nv[i] = v; s.q[i] = v * INV_N;
  }
  for (int i = tid; i < E_; i += THREADS) { s.ef[i] = edges_from[i]; s.et[i] = edges_to[i]; }
  for (int i = tid; i < E_*A_; i += THREADS) { s.mf[i] = 0.f; s.mb[i] = 0.f; }
  if (tid == 0) s.qmax = -__builtin_inff();
  __syncthreads();

  auto argmax_eval = [&]() {
    if (tid < N_) {                       // first-max argmax, matches jnp.argmax
      const float* row = &s.q[tid*A_];
      int best = 0; float bv = row[0];
#pragma unroll
      for (int a = 1; a < A_; ++a) { float v = row[a]; if (v > bv) { bv = v; best = a; } }
      s.a_cur[tid] = best;
    }
    __syncthreads();
    float part = 0.f;
    for (int n = tid; n < N_; n += THREADS) part += s.nv[n*A_ + s.a_cur[n]] * INV_N;
    for (int e = tid; e < E_; e += THREADS) {
      int af = s.a_cur[s.ef[e]], at = s.a_cur[s.et[e]];
      part += eg[(size_t)e*(A_*A_) + af*A_ + at] * INV_E;
    }
    s.red[tid] = part;
    __syncthreads();
#pragma unroll
    for (int st = THREADS/2; st > 0; st >>= 1) {
      if (tid < st) s.red[tid] += s.red[tid + st];
      __syncthreads();
    }
    if (tid == 0) {
      float qv = s.red[0];
      s.upd = (qv > s.qmax) ? 1 : 0;
      if (s.upd) s.qmax = qv;
    }
    __syncthreads();
    if (s.upd && tid < N_) s.a_best[tid] = s.a_cur[tid];
    __syncthreads();
  };

  argmax_eval();   // argmax(q0) == argmax(node_vals); evaluates initial greedy action

  for (int it = 0; it < ITERS_; ++it) {
    async_tile(eg, s.ebuf[0], tid);                       // prime pipeline
    for (int t = 0; t < NTILES_; ++t) {
      if (t + 1 < NTILES_) {
        async_tile(eg + (size_t)(t+1)*TE_*(A_*A_), s.ebuf[(t+1)&1], tid);
        asm volatile("s_wait_asynccnt 0x8" ::: "memory"); // tile t done, t+1 in flight
      } else {
        asm volatile("s_wait_asynccnt 0x0" ::: "memory");
      }
      __syncthreads();
      const float* EB = s.ebuf[t & 1];
      const int e0 = t * TE_;

      // stage vf = q[from]-mb, vb = q[to]-mf (old messages, per edge)
      for (int i = tid; i < TE_*A_; i += THREADS) {
        int le = i >> 4, a = i & 15, e = e0 + le;
        s.vf[i] = s.q[s.ef[e]*A_ + a] - s.mb[e*A_ + a];
        s.vb[i] = s.q[s.et[e]*A_ + a] - s.mf[e*A_ + a];
      }
      __syncthreads();

      // max-plus products: sf[e][at] = max_af vf[af]+M[af][at]/E ; sb[e][af] = max_at vb[at]+M[af][at]/E
      for (int i = tid; i < TE_*A_; i += THREADS) {
        int le = i >> 4, a = i & 15;
        const float* M   = EB + le*(A_*A_);
        const float* vfr = s.vf + le*A_;
        const float* vbr = s.vb + le*A_;
        float m1 = -__builtin_inff(), m2 = -__builtin_inff();
#pragma unroll
        for (int k = 0; k < A_; ++k) {
          m1 = fmaxf(m1, fmaf(M[k*A_ + a], INV_E, vfr[k]));
          m2 = fmaxf(m2, fmaf(M[a*A_ + k], INV_E, vbr[k]));
        }
        s.sf[i] = m1;
        s.sb[i] = m2;
      }
      __syncthreads();

      // mean-centering via WMMA: one wave per 16-edge block, writes into mf/mb
      {
        const int wave = tid >> 5;
        if (wave < 2)       center16(&s.sf[wave*16*A_],     &s.mf[(e0 + wave*16)*A_]);
        else if (wave < 4)  center16(&s.sb[(wave-2)*16*A_], &s.mb[(e0 + (wave-2)*16)*A_]);
      }
      __syncthreads();
    }

    // q = q0 + scatter-add(mf -> to, mb -> from) via LDS float atomics
    for (int i = tid; i < N_*A_; i += THREADS) s.q[i] = s.nv[i] * INV_N;
    __syncthreads();
    for (int i = tid; i < E_*A_; i += THREADS) {
      int e = i >> 4, a = i & 15;
      atomicAdd(&s.q[s.et[e]*A_ + a], s.mf[i]);
      atomicAdd(&s.q[s.ef[e]*A_ + a], s.mb[i]);
    }
    __syncthreads();

    argmax_eval();
  }

  // outputs: q_max (B floats) then a_max (B*N, stored as float values)
  if (tid == 0) out[b] = s.qmax;
  if (tid < N_) out[B_ + b*N_ + tid] = (float)s.a_best[tid];
}

extern "C" void kernel_launch(void* const* d_in, const int* in_sizes, int n_in,
                              void* d_out, int out_size, void* d_ws, size_t ws_size,
                              hipStream_t stream) {
  (void)in_sizes; (void)n_in; (void)out_size; (void)d_ws; (void)ws_size;
  const float* node_vals = (const float*)d_in[0];
  const float* edge_vals = (const float*)d_in[1];
  const int*   ef        = (const int*)d_in[2];
  const int*   et        = (const int*)d_in[3];
  float* out = (float*)d_out;

  hipFuncSetAttribute((const void*)dcg_maxsum_kernel,
                      hipFuncAttributeMaxDynamicSharedMemorySize, (int)sizeof(Smem));
  dcg_maxsum_kernel<<<dim3(B_), dim3(THREADS), sizeof(Smem), stream>>>(
      node_vals, edge_vals, ef, et, out);
}